// Net_48782238547968
// MI455X (gfx1250) — compile-verified
//
#include <hip/hip_runtime.h>

typedef __attribute__((ext_vector_type(2))) float v2f;
typedef __attribute__((ext_vector_type(8))) float v8f;

#define T_LEN 4096
#define B_SZ  2048
#define NTILE 4                       // 16-batch WMMA tiles per wave (64 batches)
#define LOG2E 1.4426950408889634f

__device__ __forceinline__ float fsig(float x) {        // v_exp_f32 + v_rcp_f32
    float e = __builtin_amdgcn_exp2f(-x * LOG2E);
    return __builtin_amdgcn_rcpf(1.0f + e);
}
__device__ __forceinline__ float ftanh(float x) {
    float e = __builtin_amdgcn_exp2f(x * (-2.0f * LOG2E));
    return fmaf(2.0f, __builtin_amdgcn_rcpf(1.0f + e), -1.0f);
}

// One wave owns 4 x 16 batch rows for the whole T=4096 recurrence.
// Per step:
//   per tile: gates(16x16) = WMMA(x rank-1) + WMMA(h @ w_hh^T)  [v_wmma_f32_16x16x4_f32]
//   raw gates scattered to LDS with a column swizzle so (i,f,g,o) of one
//     (tile, hidden) pair are contiguous -> ds_load_b128 gives a packed float4
//   packed update: all 32 lanes useful (lane p = tile (p&15)>>2, hidden p&3);
//     c state is 8 regs TOTAL for all 4 tiles
//   h written back to LDS in A-layout order; per-tile reload feeds next WMMA
//   fused FC: per-lane dot + shfl_xor(16) + coalesced 16-lane store
__global__ __launch_bounds__(32)
void lstm_wmma_kernel(const float* __restrict__ x,
                      const float* __restrict__ w_ih,
                      const float* __restrict__ w_hh,
                      const float* __restrict__ b_ih,
                      const float* __restrict__ b_hh,
                      const float* __restrict__ w_fc,
                      const float* __restrict__ b_fc,
                      float* __restrict__ out)
{
    const int L    = threadIdx.x;          // 0..31
    const int n    = L & 15;               // C-layout column; batch row for x loads
    const int half = L >> 4;               // C-layout row half / A-B K half
    const int base = blockIdx.x * (16 * NTILE);

    // ---- WMMA operands (shared by all tiles) ----
    v2f bW;                                // B[k][n] = w_hh[n][k], K pair by half
    bW.x = w_hh[n * 4 + 2 * half];
    bW.y = w_hh[n * 4 + 2 * half + 1];
    v2f bX;                                // input projection: row K=0 = w_ih[n]
    bX.x = half ? 0.0f : w_ih[n];
    bX.y = 0.0f;
    const float bias_n = b_ih[n] + b_hh[n];
    v8f biasC;
#pragma unroll
    for (int v = 0; v < 8; ++v) biasC[v] = bias_n;

    // ---- per-lane LDS address bases ----
    // gate scatter (C layout -> swizzled): column c=L&15 goes to slot (c&3)*4+(c>>2)
    const int sw  = (L & 3) * 4 + ((L >> 2) & 3) + half * 16;      // + u*256 + v*32
    // packed gather: lane p = (tile (p&15)>>2, hidden p&3), row = v + 8*(p>>4)
    const int up  = (n >> 2);
    const int j   = L & 3;
    const int pa  = up * 256 + half * 16 + j * 4;                  // + v*32 (float4)
    // packed h store in A-layout order: addr = u*64 + j*16 + row
    const int hb  = up * 64 + j * 16 + half * 8;                   // + v
    // FC weights per half (A layout: lane holds K=2*half, 2*half+1)
    const int jlo = half * 2;
    const float wf0 = w_fc[jlo];
    const float wf1 = w_fc[jlo + 1];
    const float bf  = b_fc[0];

    v2f hA[NTILE];
#pragma unroll
    for (int u = 0; u < NTILE; ++u) { hA[u].x = 0.0f; hA[u].y = 0.0f; }
    float c[8];                            // packed c for ALL 4 tiles
#pragma unroll
    for (int v = 0; v < 8; ++v) c[v] = 0.0f;

    __shared__ __align__(16) float ldsA[NTILE * 256]; // swizzled gates
    __shared__ float ldsH[NTILE * 64];                // h in A-layout order

    const float* xq = x + base + n;
    float*       oq = out + base + n;

    float xv[NTILE];
#pragma unroll
    for (int u = 0; u < NTILE; ++u) xv[u] = xq[u * 16];

    for (int t = 0; t < T_LEN; ++t) {
        // software pipeline: next-step loads + deep prefetch first
        float xnext[NTILE];
#pragma unroll
        for (int u = 0; u < NTILE; ++u)
            xnext[u] = (t + 1 < T_LEN) ? xq[(size_t)(t + 1) * B_SZ + u * 16] : 0.0f;
        __builtin_prefetch(xq + (size_t)(t + 16) * B_SZ, 0, 3);

        // gates = x*w_ih^T + bias, then += h @ w_hh^T (independent per tile),
        // then scatter raw gates to LDS (swizzled columns)
        v8f g[NTILE];
#pragma unroll
        for (int u = 0; u < NTILE; ++u) {
            v2f aX;
            aX.x = half ? 0.0f : xv[u];
            aX.y = 0.0f;
            g[u] = __builtin_amdgcn_wmma_f32_16x16x4_f32(
                false, aX, false, bX, (short)0, biasC, false, false);
            g[u] = __builtin_amdgcn_wmma_f32_16x16x4_f32(
                false, hA[u], false, bW, (short)0, g[u], false, false);
        }
#pragma unroll
        for (int u = 0; u < NTILE; ++u)
#pragma unroll
            for (int v = 0; v < 8; ++v)
                ldsA[u * 256 + v * 32 + sw] = g[u][v];
        __syncthreads();                   // single-wave WG: near-free

        // packed update: every lane owns one (tile, hidden) pair, 8 rows in v
#pragma unroll
        for (int v = 0; v < 8; ++v) {
            float4 q = *(const float4*)&ldsA[pa + v * 32];   // (i, f, g, o)
            float iv = fsig(q.x);
            float fv = fsig(q.y);
            float gv = ftanh(q.z);
            float ov = fsig(q.w);
            c[v] = fmaf(fv, c[v], iv * gv);
            ldsH[hb + v] = ov * ftanh(c[v]);
        }
        __syncthreads();

        // reload h in A layout + fused FC output
#pragma unroll
        for (int u = 0; u < NTILE; ++u) {
            float h0 = ldsH[u * 64 + jlo * 16 + n];
            float h1 = ldsH[u * 64 + jlo * 16 + 16 + n];
            hA[u].x = h0;
            hA[u].y = h1;
            float p = fmaf(h0, wf0, h1 * wf1);
            p += __shfl_xor(p, 16, 32);
            if (half == 0)
                oq[(size_t)t * B_SZ + u * 16] = p + bf;
        }

#pragma unroll
        for (int u = 0; u < NTILE; ++u) xv[u] = xnext[u];
    }
}

extern "C" void kernel_launch(void* const* d_in, const int* in_sizes, int n_in,
                              void* d_out, int out_size, void* d_ws, size_t ws_size,
                              hipStream_t stream) {
    const float* x    = (const float*)d_in[0];
    const float* w_ih = (const float*)d_in[1];
    const float* w_hh = (const float*)d_in[2];
    const float* b_ih = (const float*)d_in[3];
    const float* b_hh = (const float*)d_in[4];
    const float* w_fc = (const float*)d_in[5];
    const float* b_fc = (const float*)d_in[6];
    float* out = (float*)d_out;

    lstm_wmma_kernel<<<dim3(B_SZ / (16 * NTILE)), dim3(32), 0, stream>>>(
        x, w_ih, w_hh, b_ih, b_hh, w_fc, b_fc, out);
}